// ContourRepairSDF_41540923687070
// MI455X (gfx1250) — compile-verified
//
#include <hip/hip_runtime.h>
#include <math.h>

#define N_Q    131072
#define PC_K   2048
#define RADIUS 0.5f

typedef __attribute__((ext_vector_type(2))) float v2f;
typedef __attribute__((ext_vector_type(8))) float v8f;

// ---- workspace layout (uint32 indices into d_ws) ----
//  [0,2048)  : histogram of |dist| float-bit top-11-bits
//  2048      : ctrA (strict-below compaction counter)
//  2049      : ctrB (boundary-bin counter)
//  2050      : threshold bin
//  2051      : base_below (count strictly below threshold bin)
//  2052      : remaining (slots to fill from boundary bin)
//  byte 8448 : float4 reps[2048]  = (-2x, -2y, -2z, r^2)
#define WS_CTRA   2048
#define WS_CTRB   2049
#define WS_TBIN   2050
#define WS_BASE   2051
#define WS_REM    2052
#define WS_REPS_BYTES 8448

#define FLT_MAX_BITS 0x7f7fffff

// Integer min on raw f32 bits: exact for non-negative floats; tiny-negative
// d^2 (rounding noise) wins and is clamped to 0 in the epilogue, matching the
// reference's max(d2,0). Integer smin gets no canonicalize and reassociates
// freely -> v_min3_i32 trees.
__device__ __forceinline__ int imin2(int a, int b) { return a < b ? a : b; }

__global__ void k_init(unsigned* ws) {
    for (int i = threadIdx.x; i < 2056; i += 256) ws[i] = 0u;
}

__device__ __forceinline__ float sdf_dist(const float* __restrict__ q, int i,
                                          float& x, float& y, float& z) {
    x = q[3 * i + 0];
    y = q[3 * i + 1];
    z = q[3 * i + 2];
    return sqrtf(x * x + y * y + z * z) - RADIUS;
}

__global__ void k_hist(const float* __restrict__ q, unsigned* __restrict__ ws) {
    int i = blockIdx.x * blockDim.x + threadIdx.x;
    float x, y, z;
    float a = fabsf(sdf_dist(q, i, x, y, z));
    unsigned bin = __float_as_uint(a) >> 20;   // sign==0 -> 11 bits, monotone in a
    atomicAdd(&ws[bin], 1u);
}

__global__ void k_scan(unsigned* __restrict__ ws) {
    unsigned cum = 0, tb = 2047, base = 0, rem = 0;
    bool done = false;
    for (int b = 0; b < 2048; ++b) {
        unsigned c = ws[b];
        if (!done && cum + c >= PC_K) {
            tb = (unsigned)b; base = cum; rem = PC_K - cum; done = true;
        }
        if (!done) cum += c;
    }
    if (!done) { tb = 2047u; base = cum; rem = 0u; }
    ws[WS_TBIN] = tb; ws[WS_BASE] = base; ws[WS_REM] = rem;
}

__global__ void k_compact(const float* __restrict__ q, unsigned* __restrict__ ws,
                          float4* __restrict__ reps) {
    int i = blockIdx.x * blockDim.x + threadIdx.x;
    float x, y, z;
    float a = fabsf(sdf_dist(q, i, x, y, z));
    unsigned bin = __float_as_uint(a) >> 20;
    unsigned tb  = ws[WS_TBIN];
    int idx = -1;
    if (bin < tb) {
        idx = (int)atomicAdd(&ws[WS_CTRA], 1u);
    } else if (bin == tb) {
        unsigned j = atomicAdd(&ws[WS_CTRB], 1u);
        if (j < ws[WS_REM]) idx = (int)(ws[WS_BASE] + j);
    }
    if (idx >= 0) {
        float r2 = x * x + y * y + z * z;
        reps[idx] = make_float4(-2.0f * x, -2.0f * y, -2.0f * z, r2);
    }
}

// One wave handles 16 query rows; 128 chunks of 16 reps, one f32 WMMA per chunk.
// D[M,N] = A(16x4) * B(4x16) + C,  A=[qx qy qz 1], B=[-2rx;-2ry;-2rz;r2], C=q2[M]
// LDS holds B in exact wave layout: chunk c -> 32 float2s, [0..15]=(K0,K1) for
// lanes 0-15, [16..31]=(K2,K3) for lanes 16-31, so the hot loop needs no selects.
__global__ __launch_bounds__(256) void k_main(const float* __restrict__ q,
                                              const float4* __restrict__ reps,
                                              float* __restrict__ out) {
    __shared__ float2 srep2[PC_K * 2];          // 32 KB of the 320 KB WGP pool
    int t = threadIdx.x;
    for (int k = t; k < PC_K; k += 256) {
        float4 f = reps[k];
        int c = k >> 4, j = k & 15;
        srep2[c * 32 + j]      = make_float2(f.x, f.y);   // K0,K1 = -2rx,-2ry
        srep2[c * 32 + 16 + j] = make_float2(f.z, f.w);   // K2,K3 = -2rz, r^2
    }
    __syncthreads();

    int lane = t & 31;
    int wave = t >> 5;
    int m0   = blockIdx.x * 128 + wave * 16;
    int row  = m0 + (lane & 15);

    float x = q[3 * row + 0], y = q[3 * row + 1], z = q[3 * row + 2];
    float q2 = x * x + y * y + z * z;
    bool hi = (lane & 16) != 0;                  // lanes 16..31 hold K=2,3 of A

    v2f A;
    A.x = hi ? z : x;                            // K0 / K2
    A.y = hi ? 1.0f : y;                         // K1 / K3 (pad with 1 for +r^2 term)

    // C[v] = q2 of row (v + 8*hi): gather other lanes' q2 via bpermute
    int sbase = hi ? 8 : 0;
    v8f C;
#pragma unroll
    for (int v = 0; v < 8; ++v) C[v] = __shfl(q2, sbase + v, 32);

    int dmin[8];
#pragma unroll
    for (int v = 0; v < 8; ++v) dmin[v] = FLT_MAX_BITS;

#pragma unroll 4
    for (int c = 0; c < PC_K / 16; ++c) {
        float2 b2 = srep2[c * 32 + lane];        // ds_load_b64, conflict-free
        v2f B; B.x = b2.x; B.y = b2.y;
        v8f D = __builtin_amdgcn_wmma_f32_16x16x4_f32(
            /*neg_a=*/false, A, /*neg_b=*/false, B,
            /*c_mod=*/(short)0, C, /*reuse_a=*/false, /*reuse_b=*/false);
#pragma unroll
        for (int v = 0; v < 8; ++v)
            dmin[v] = imin2(dmin[v], __float_as_int(D[v]));
    }

    // row-min across the 16 lanes of each half (xor masks stay within a half)
#pragma unroll
    for (int mask = 8; mask >= 1; mask >>= 1) {
#pragma unroll
        for (int v = 0; v < 8; ++v)
            dmin[v] = imin2(dmin[v], __shfl_xor(dmin[v], mask, 32));
    }

    if ((lane & 15) == 0) {         // lane 0 -> rows m0..m0+7, lane 16 -> m0+8..m0+15
        int rbase = m0 + (hi ? 8 : 0);
#pragma unroll
        for (int v = 0; v < 8; ++v) {
            int m = rbase + v;
            float xx = q[3 * m + 0], yy = q[3 * m + 1], zz = q[3 * m + 2];
            float dist = sqrtf(xx * xx + yy * yy + zz * zz) - RADIUS;
            float d2 = fmaxf(__int_as_float(dmin[v]), 0.0f);
            float nd = (d2 > 0.0f) ? sqrtf(d2) : 0.0f;
            float sg = (dist > 0.0f) ? 1.0f : ((dist < 0.0f) ? -1.0f : 0.0f);
            out[m] = fminf(nd * sg, dist);
        }
    }
}

extern "C" void kernel_launch(void* const* d_in, const int* in_sizes, int n_in,
                              void* d_out, int out_size, void* d_ws, size_t ws_size,
                              hipStream_t stream) {
    const float* q = (const float*)d_in[0];
    float* out = (float*)d_out;
    unsigned* ws = (unsigned*)d_ws;
    float4* reps = (float4*)((char*)d_ws + WS_REPS_BYTES);

    k_init   <<<1,   256, 0, stream>>>(ws);
    k_hist   <<<512, 256, 0, stream>>>(q, ws);
    k_scan   <<<1,   1,   0, stream>>>(ws);
    k_compact<<<512, 256, 0, stream>>>(q, ws, reps);
    k_main   <<<1024,256, 0, stream>>>(q, reps, out);
}